// PointNet2FeaExtractor_12850542149710
// MI455X (gfx1250) — compile-verified
//
#include <hip/hip_runtime.h>

#define BN_EPS 1e-5f

typedef __attribute__((ext_vector_type(2))) float v2f;
typedef __attribute__((ext_vector_type(8))) float v8f;
typedef unsigned int u32x4 __attribute__((ext_vector_type(4)));
typedef int i32x4 __attribute__((ext_vector_type(4)));
typedef int i32x8 __attribute__((ext_vector_type(8)));

#ifndef __has_builtin
#define __has_builtin(x) 0
#endif
#if __has_builtin(__builtin_amdgcn_tensor_load_to_lds)
#define HAVE_TDM 1
#else
#define HAVE_TDM 0
#endif

// ============================================================================
// WMMA fp32 GEMM:  Y[M,O] = X[M,K] * W[O,K]^T   (einsum '...c,oc->...o')
// PRECONDITION: M % 16 == 0, O % 16 == 0 (guaranteed by caller); grid exact.
// One wave computes one 16x16 tile via V_WMMA_F32_16X16X4_F32.
// The 16xK X-panel is staged into LDS by the Tensor Data Mover (one
// tensor_load_to_lds per wave), then A-fragments are read from LDS.
// A frag (16x4 f32): lane l<16 vgpr j -> A[l][j];  lane l>=16 -> A[l-16][j+2]
// B frag (4x16 f32): lane l<16 vgpr j -> B[j][l];  lane l>=16 -> B[j+2][l-16]
// D (16x16 f32): vgpr r, lane l<16 -> D[r][l]; lane l>=16 -> D[r+8][l-16]
// ============================================================================
#define GEMM_KMAX 68   // max K over all call sites is 67

__global__ __launch_bounds__(32) void gemm_wmma_f32(const float* __restrict__ X,
                                                    const float* __restrict__ W,
                                                    float* __restrict__ Y,
                                                    int M, int K, int O)
{
    __shared__ float atile[16 * GEMM_KMAX];

    const int lane = threadIdx.x & 31;
    const int half = lane >> 4;    // 0 or 1
    const int l16  = lane & 15;
    const int tm   = blockIdx.x;
    const int to   = blockIdx.y;

#if HAVE_TDM
    // ---- Tensor Data Mover: copy 16 x K fp32 panel of X into LDS ----------
    {
        const unsigned long long ga =
            (unsigned long long)(size_t)(X + (size_t)tm * 16 * K);
        const unsigned laddr = (unsigned)(size_t)(void*)atile;  // LDS byte addr

        u32x4 g0;
        g0[0] = 1u;                                   // count=1, no gather
        g0[1] = laddr;                                // lds_addr
        g0[2] = (unsigned)(ga & 0xffffffffu);         // global_addr[31:0]
        g0[3] = (unsigned)((ga >> 32) & 0x1ffffffu)   // global_addr[56:32]
                | (2u << 30);                         // type=2 ("image")

        i32x8 g1;
        g1[0] = (2 << 16);                            // data_size=4B, mask=0
        g1[1] = (K & 0xffff) << 16;                   // tensor_dim0[15:0]
        g1[2] = (16 << 16) | ((K >> 16) & 0xffff);    // tdim0[31:16]=0|tdim1 lo=16
        g1[3] = (K & 0xffff) << 16;                   // tdim1 hi=0 | tile_dim0=K
        g1[4] = 16;                                   // tile_dim1=16, tile_dim2=0
        g1[5] = K;                                    // tensor_dim0_stride[31:0]
        g1[6] = 0;                                    // stride hi / dim1_stride lo
        g1[7] = 0;

        i32x4 g2 = {0, 0, 0, 0};
        i32x4 g3 = {0, 0, 0, 0};
#if __clang_major__ >= 23
        i32x8 g4 = {0, 0, 0, 0, 0, 0, 0, 0};
        __builtin_amdgcn_tensor_load_to_lds(g0, g1, g2, g3, g4, 0);
#else
        __builtin_amdgcn_tensor_load_to_lds(g0, g1, g2, g3, 0);
#endif
#if __has_builtin(__builtin_amdgcn_s_wait_tensorcnt)
        __builtin_amdgcn_s_wait_tensorcnt(0);
#else
        asm volatile("s_wait_tensorcnt 0x0" ::: "memory");
#endif
        asm volatile("" ::: "memory");   // LDS tile now valid; block reordering
    }
    const float* __restrict__ ar = atile + l16 * K;          // A row (LDS)
#else
    const float* __restrict__ ar = X + (size_t)(tm * 16 + l16) * K;  // A row
#endif
    const float* __restrict__ wr = W + (size_t)(to * 16 + l16) * K;  // B row

    v8f acc = {0.f, 0.f, 0.f, 0.f, 0.f, 0.f, 0.f, 0.f};

    const int K4 = K & ~3;
    for (int k0 = 0; k0 < K4; k0 += 4) {
        const int ka = k0 + 2 * half;
        v2f a, b;
        a.x = ar[ka + 0];
        a.y = ar[ka + 1];
        b.x = wr[ka + 0];
        b.y = wr[ka + 1];
        acc = __builtin_amdgcn_wmma_f32_16x16x4_f32(false, a, false, b,
                                                    (short)0, acc, false, false);
    }
    if (K4 < K) {       // ragged-K tail: clamped addresses + 0/1 mask (no exec)
        const int ka  = K4 + 2 * half;
        const int k0c = (ka + 0 < K) ? ka + 0 : K - 1;
        const int k1c = (ka + 1 < K) ? ka + 1 : K - 1;
        const float m0 = (ka + 0 < K) ? 1.f : 0.f;
        const float m1 = (ka + 1 < K) ? 1.f : 0.f;
        v2f a, b;
        a.x = ar[k0c] * m0;
        a.y = ar[k1c] * m1;
        b.x = wr[k0c] * m0;
        b.y = wr[k1c] * m1;
        acc = __builtin_amdgcn_wmma_f32_16x16x4_f32(false, a, false, b,
                                                    (short)0, acc, false, false);
    }

    float* __restrict__ yp = Y + (size_t)(tm * 16 + 8 * half) * O + to * 16 + l16;
#pragma unroll
    for (int r = 0; r < 8; ++r) yp[(size_t)r * O] = acc[r];
}

// ============================================================================
// Farthest point sampling: one block per batch. dist[] lives in LDS.
// ============================================================================
#define FPS_MAX_N 8192
#define FPS_T 512

__global__ __launch_bounds__(FPS_T) void fps_kernel(const float* __restrict__ xyz,
                                                    int N, int npoint,
                                                    int* __restrict__ idx_out)
{
    __shared__ float dist[FPS_MAX_N];
    __shared__ float rv[FPS_T];
    __shared__ int   ri[FPS_T];

    const int b = blockIdx.x;
    const int t = threadIdx.x;
    const float* base = xyz + (size_t)b * N * 3;

    for (int i = t; i < N; i += FPS_T) dist[i] = 1e10f;
    __syncthreads();

    int far = 0;
    for (int s = 0; s < npoint; ++s) {
        if (t == 0) idx_out[b * npoint + s] = far;
        const float cx = base[far * 3 + 0];
        const float cy = base[far * 3 + 1];
        const float cz = base[far * 3 + 2];

        float bv = -1.f; int bi = 0x7fffffff;
        for (int i = t; i < N; i += FPS_T) {
            const float dx = base[i * 3 + 0] - cx;
            const float dy = base[i * 3 + 1] - cy;
            const float dz = base[i * 3 + 2] - cz;
            const float d  = dx * dx + dy * dy + dz * dz;
            const float nd = fminf(dist[i], d);
            dist[i] = nd;
            if (nd > bv) { bv = nd; bi = i; }   // strict > keeps lowest index
        }
        rv[t] = bv; ri[t] = bi;
        __syncthreads();
        for (int off = FPS_T / 2; off > 0; off >>= 1) {
            if (t < off) {
                if (rv[t + off] > rv[t] ||
                    (rv[t + off] == rv[t] && ri[t + off] < ri[t])) {
                    rv[t] = rv[t + off]; ri[t] = ri[t + off];
                }
            }
            __syncthreads();
        }
        far = ri[0];
        __syncthreads();
    }
}

// gather new_xyz = xyz[b, idx[b,s], :]
__global__ void gather3(const float* __restrict__ xyz, const int* __restrict__ idx,
                        float* __restrict__ out, int N, int S, int total)
{
    const int i = blockIdx.x * blockDim.x + threadIdx.x;
    if (i >= total) return;
    const int b = i / S;
    const int j = idx[i];
    const float* p = xyz + ((size_t)b * N + j) * 3;
    out[(size_t)i * 3 + 0] = p[0];
    out[(size_t)i * 3 + 1] = p[1];
    out[(size_t)i * 3 + 2] = p[2];
}

// ============================================================================
// Fused ball-query + grouping. One thread per query point.
// feat_first=0: out = [gxyz(3), feat(Cf)] ; feat_first=1: out = [feat(Cf), gpos(3)]
// ============================================================================
__global__ void ball_group(const float* __restrict__ xyz, const float* __restrict__ feat,
                           const float* __restrict__ new_xyz, float* __restrict__ out,
                           int N, int S, int nsample, int Cf, float r2,
                           int feat_first, int total)
{
    const int gid = blockIdx.x * blockDim.x + threadIdx.x;
    if (gid >= total) return;
    const int b = gid / S;
    const float* px = xyz + (size_t)b * N * 3;

    const float nx = new_xyz[(size_t)gid * 3 + 0];
    const float ny = new_xyz[(size_t)gid * 3 + 1];
    const float nz = new_xyz[(size_t)gid * 3 + 2];

    int nb[32];
    int cnt = 0;
    for (int j = 0; j < N && cnt < nsample; ++j) {
        if ((j & 31) == 0 && j + 32 < N)
            __builtin_prefetch(&px[(j + 32) * 3], 0, 0);   // global_prefetch_b8
        const float dx = px[j * 3 + 0] - nx;
        const float dy = px[j * 3 + 1] - ny;
        const float dz = px[j * 3 + 2] - nz;
        if (dx * dx + dy * dy + dz * dz <= r2) nb[cnt++] = j;
    }
    const int pad = (cnt > 0) ? nb[0] : 0;

    const int C = Cf + 3;
    float* o = out + (size_t)gid * nsample * C;
    for (int k = 0; k < nsample; ++k) {
        const int j = (k < cnt) ? nb[k] : pad;
        const float gx = px[j * 3 + 0] - nx;
        const float gy = px[j * 3 + 1] - ny;
        const float gz = px[j * 3 + 2] - nz;
        const float* f = feat + ((size_t)b * N + j) * Cf;
        float* ok = o + (size_t)k * C;
        if (feat_first) {
            for (int c = 0; c < Cf; ++c) ok[c] = f[c];
            ok[Cf + 0] = gx; ok[Cf + 1] = gy; ok[Cf + 2] = gz;
        } else {
            ok[0] = gx; ok[1] = gy; ok[2] = gz;
            for (int c = 0; c < Cf; ++c) ok[3 + c] = f[c];
        }
    }
}

// ============================================================================
// Global batch-norm statistics (two-pass) + apply
// ============================================================================
__global__ void zero_stats(float* __restrict__ st, int n)
{
    const int i = threadIdx.x;
    if (i < n) st[i] = 0.f;
}

__global__ __launch_bounds__(256) void bn_stats(const float* __restrict__ x,
                                                float* __restrict__ sums,
                                                float* __restrict__ sqs,
                                                long long M, int O)
{
    __shared__ float ssum[64];
    __shared__ float ssq[64];
    const int t = threadIdx.x;
    if (t < 64) { ssum[t] = 0.f; ssq[t] = 0.f; }
    __syncthreads();
    const long long total  = M * (long long)O;
    const long long stride = (long long)gridDim.x * blockDim.x;
    for (long long i = (long long)blockIdx.x * blockDim.x + t; i < total; i += stride) {
        const float v = x[i];
        const int o = (int)(i % O);
        atomicAdd(&ssum[o], v);
        atomicAdd(&ssq[o], v * v);
    }
    __syncthreads();
    if (t < O) {
        atomicAdd(&sums[t], ssum[t]);
        atomicAdd(&sqs[t],  ssq[t]);
    }
}

__global__ void bn_relu_apply(float* __restrict__ x, const float* __restrict__ sums,
                              const float* __restrict__ sqs, const float* __restrict__ g,
                              const float* __restrict__ bb, long long M, int O)
{
    const long long i = (long long)blockIdx.x * blockDim.x + threadIdx.x;
    if (i >= M * (long long)O) return;
    const int o = (int)(i % O);
    const float mu  = sums[o] / (float)M;
    const float var = sqs[o] / (float)M - mu * mu;
    const float v = (x[i] - mu) * rsqrtf(var + BN_EPS) * g[o] + bb[o];
    x[i] = v > 0.f ? v : 0.f;
}

// max over nsample axis: x[R, ns, O] -> y[R, O]
__global__ void maxpool_kernel(const float* __restrict__ x, float* __restrict__ y,
                               long long R, int ns, int O)
{
    const long long i = (long long)blockIdx.x * blockDim.x + threadIdx.x;
    if (i >= R * (long long)O) return;
    const long long r = i / O;
    const int o = (int)(i % O);
    const float* p = x + ((size_t)r * ns) * O + o;
    float m = -3.4e38f;
    for (int k = 0; k < ns; ++k) m = fmaxf(m, p[(size_t)k * O]);
    y[i] = m;
}

// y[R, Ca+Cb] = concat(a[R,Ca], b[R,Cb])
__global__ void concat2(const float* __restrict__ a, int Ca,
                        const float* __restrict__ b, int Cb,
                        float* __restrict__ y, long long R)
{
    const int C = Ca + Cb;
    const long long i = (long long)blockIdx.x * blockDim.x + threadIdx.x;
    if (i >= R * (long long)C) return;
    const long long r = i / C;
    const int c = (int)(i % C);
    y[i] = (c < Ca) ? a[r * Ca + c] : b[r * Cb + (c - Ca)];
}

// ============================================================================
// feature_prop: 3-NN inverse-distance interpolation + concat original features
// out[b,n] = [ interp(C2), feat1(C1) ]
// ============================================================================
__global__ void three_interp(const float* __restrict__ pos1, const float* __restrict__ pos2,
                             const float* __restrict__ feat2, const float* __restrict__ feat1,
                             float* __restrict__ out, int N1, int N2, int C2, int C1,
                             int total)
{
    const int gid = blockIdx.x * blockDim.x + threadIdx.x;
    if (gid >= total) return;
    const int b = gid / N1;

    const float px = pos1[(size_t)gid * 3 + 0];
    const float py = pos1[(size_t)gid * 3 + 1];
    const float pz = pos1[(size_t)gid * 3 + 2];
    const float* p2 = pos2 + (size_t)b * N2 * 3;

    float d0 = 3e38f, d1 = 3e38f, d2 = 3e38f;
    int   i0 = 0,     i1 = 0,     i2 = 0;
    for (int j = 0; j < N2; ++j) {
        const float dx = p2[j * 3 + 0] - px;
        const float dy = p2[j * 3 + 1] - py;
        const float dz = p2[j * 3 + 2] - pz;
        const float d  = dx * dx + dy * dy + dz * dz;
        if (d < d2) {
            if (d < d0)      { d2 = d1; i2 = i1; d1 = d0; i1 = i0; d0 = d; i0 = j; }
            else if (d < d1) { d2 = d1; i2 = i1; d1 = d;  i1 = j; }
            else             { d2 = d;  i2 = j; }
        }
    }
    d0 = fmaxf(d0, 1e-10f); d1 = fmaxf(d1, 1e-10f); d2 = fmaxf(d2, 1e-10f);
    float w0 = 1.f / d0, w1 = 1.f / d1, w2 = 1.f / d2;
    const float sw = w0 + w1 + w2;
    w0 /= sw; w1 /= sw; w2 /= sw;

    const float* f2 = feat2 + (size_t)b * N2 * C2;
    float* o = out + (size_t)gid * (C2 + C1);
    for (int c = 0; c < C2; ++c)
        o[c] = w0 * f2[(size_t)i0 * C2 + c] + w1 * f2[(size_t)i1 * C2 + c]
             + w2 * f2[(size_t)i2 * C2 + c];
    const float* f1 = feat1 + (size_t)gid * C1;
    for (int c = 0; c < C1; ++c) o[C2 + c] = f1[c];
}

// ============================================================================
// final head: bn_relu(x; bn1) then 32->12 linear + bias, concat with xyz -> 15ch
// ============================================================================
__global__ void final_head(const float* __restrict__ x, const float* __restrict__ sums,
                           const float* __restrict__ sqs, const float* __restrict__ g,
                           const float* __restrict__ bb, const float* __restrict__ W,
                           const float* __restrict__ wb, const float* __restrict__ pts,
                           float* __restrict__ out, long long M, int O)
{
    const long long i = (long long)blockIdx.x * blockDim.x + threadIdx.x;
    if (i >= M) return;
    float xn[32];
    for (int o = 0; o < O; ++o) {
        const float mu  = sums[o] / (float)M;
        const float var = sqs[o] / (float)M - mu * mu;
        const float v = (x[i * O + o] - mu) * rsqrtf(var + BN_EPS) * g[o] + bb[o];
        xn[o] = v > 0.f ? v : 0.f;
    }
    float* od = out + i * 15;
    od[0] = pts[i * 3 + 0];
    od[1] = pts[i * 3 + 1];
    od[2] = pts[i * 3 + 2];
    for (int o2 = 0; o2 < 12; ++o2) {
        float acc = wb[o2];
        for (int c = 0; c < O; ++c) acc += xn[c] * W[o2 * O + c];
        od[3 + o2] = acc;
    }
}

// ============================================================================
// Host orchestration
// ============================================================================
static inline dim3 gs(long long n, int bs) { return dim3((unsigned)((n + bs - 1) / bs)); }

extern "C" void kernel_launch(void* const* d_in, const int* in_sizes, int n_in,
                              void* d_out, int out_size, void* d_ws, size_t ws_size,
                              hipStream_t stream)
{
    (void)in_sizes; (void)n_in; (void)out_size; (void)ws_size;

    const int B = 2, N = 8192, NP1 = 4096, NP2 = 1024, NS = 32, NSU = 8;
    const float R0 = 0.02f;
    const float r1sq = (20 * R0) * (20 * R0);    // 0.16
    const float r2sq = (40 * R0) * (40 * R0);    // 0.64
    const float rusq = (120 * R0) * (120 * R0);  // 5.76

    const float* points1 = (const float*)d_in[0];
    const float* fea1    = (const float*)d_in[1];
    const float* points2 = (const float*)d_in[2];
    const float* fea2    = (const float*)d_in[3];

    enum { SA1_W1, SA1_G1, SA1_B1, SA1_W2, SA1_G2, SA1_B2,
           SA2_W1, SA2_G1, SA2_B1, SA2_W2, SA2_G2, SA2_B2,
           SU1_W1, SU1_G1, SU1_B1, SU1_W2, SU1_G2, SU1_B2,
           FP_W1,  FP_G1,  FP_B1,  BN1_G,  BN1_B,  CONV2_W, CONV2_B };
    const float* prm[25];
    for (int i = 0; i < 25; ++i) prm[i] = (const float*)d_in[4 + i];

    // ---- workspace carve (bump allocator, 256B aligned) ----
    char* ws = (char*)d_ws;
    size_t off = 0;
    auto carve = [&](size_t bytes) -> char* {
        char* p = ws + off;
        off = (off + bytes + 255) & ~(size_t)255;
        return p;
    };
    int*   fps_idx = (int*)  carve((size_t)B * NP1 * sizeof(int));
    float* l1_pc   = (float*)carve((size_t)B * NP1 * 3  * sizeof(float));
    float* l1_f    = (float*)carve((size_t)B * NP1 * 32 * sizeof(float));
    float* l2_pc   = (float*)carve((size_t)B * NP2 * 3  * sizeof(float));
    float* l2_f    = (float*)carve((size_t)B * NP2 * 64 * sizeof(float));
    float* l1_new  = (float*)carve((size_t)B * NP1 * 32 * sizeof(float));
    float* pooled  = (float*)carve((size_t)B * NP1 * 32 * sizeof(float));
    float* stats   = (float*)carve(128 * sizeof(float));     // sums[64] | sqs[64]
    float* A       = (float*)carve((size_t)9000000 * sizeof(float));
    float* Bf      = (float*)carve((size_t)9000000 * sizeof(float));

    auto conv_bn_relu = [&](const float* Xp, const float* Wp, const float* gp,
                            const float* bp, long long M, int K, int O, float* Yp) {
        dim3 grid((unsigned)(M / 16), (unsigned)(O / 16));   // M,O multiples of 16
        gemm_wmma_f32<<<grid, dim3(32), 0, stream>>>(Xp, Wp, Yp, (int)M, K, O);
        zero_stats<<<dim3(1), dim3(128), 0, stream>>>(stats, 128);
        bn_stats<<<dim3(256), dim3(256), 0, stream>>>(Yp, stats, stats + 64, M, O);
        bn_relu_apply<<<gs(M * O, 256), dim3(256), 0, stream>>>(Yp, stats, stats + 64,
                                                                gp, bp, M, O);
    };

    auto run_branch = [&](const float* pts, const float* fea, float* outp) {
        // ---- SA1: 8192 -> 4096, r=0.4, ns=32, 13->16->32 ----
        fps_kernel<<<dim3(B), dim3(FPS_T), 0, stream>>>(pts, N, NP1, fps_idx);
        gather3<<<gs((long long)B * NP1, 128), dim3(128), 0, stream>>>(
            pts, fps_idx, l1_pc, N, NP1, B * NP1);
        ball_group<<<gs((long long)B * NP1, 128), dim3(128), 0, stream>>>(
            pts, fea, l1_pc, A, N, NP1, NS, 10, r1sq, 0, B * NP1);
        conv_bn_relu(A,  prm[SA1_W1], prm[SA1_G1], prm[SA1_B1],
                     (long long)B * NP1 * NS, 13, 16, Bf);
        conv_bn_relu(Bf, prm[SA1_W2], prm[SA1_G2], prm[SA1_B2],
                     (long long)B * NP1 * NS, 16, 32, A);
        maxpool_kernel<<<gs((long long)B * NP1 * 32, 256), dim3(256), 0, stream>>>(
            A, l1_f, (long long)B * NP1, NS, 32);

        // ---- SA2: 4096 -> 1024, r=0.8, ns=32, 35->32->64 ----
        fps_kernel<<<dim3(B), dim3(FPS_T), 0, stream>>>(l1_pc, NP1, NP2, fps_idx);
        gather3<<<gs((long long)B * NP2, 128), dim3(128), 0, stream>>>(
            l1_pc, fps_idx, l2_pc, NP1, NP2, B * NP2);
        ball_group<<<gs((long long)B * NP2, 128), dim3(128), 0, stream>>>(
            l1_pc, l1_f, l2_pc, A, NP1, NP2, NS, 32, r2sq, 0, B * NP2);
        conv_bn_relu(A,  prm[SA2_W1], prm[SA2_G1], prm[SA2_B1],
                     (long long)B * NP2 * NS, 35, 32, Bf);
        conv_bn_relu(Bf, prm[SA2_W2], prm[SA2_G2], prm[SA2_B2],
                     (long long)B * NP2 * NS, 32, 64, A);
        maxpool_kernel<<<gs((long long)B * NP2 * 64, 256), dim3(256), 0, stream>>>(
            A, l2_f, (long long)B * NP2, NS, 64);

        // ---- set_upconv: 1024 -> 4096, r=2.4, ns=8, mlp1 67->32, mlp2 64->32 ----
        ball_group<<<gs((long long)B * NP1, 128), dim3(128), 0, stream>>>(
            l2_pc, l2_f, l1_pc, A, NP2, NP1, NSU, 64, rusq, 1, B * NP1);
        conv_bn_relu(A, prm[SU1_W1], prm[SU1_G1], prm[SU1_B1],
                     (long long)B * NP1 * NSU, 67, 32, Bf);
        maxpool_kernel<<<gs((long long)B * NP1 * 32, 256), dim3(256), 0, stream>>>(
            Bf, pooled, (long long)B * NP1, NSU, 32);
        concat2<<<gs((long long)B * NP1 * 64, 256), dim3(256), 0, stream>>>(
            pooled, 32, l1_f, 32, A, (long long)B * NP1);
        conv_bn_relu(A, prm[SU1_W2], prm[SU1_G2], prm[SU1_B2],
                     (long long)B * NP1, 64, 32, Bf);
        hipMemcpyAsync(l1_new, Bf, (size_t)B * NP1 * 32 * sizeof(float),
                       hipMemcpyDeviceToDevice, stream);

        // ---- feature_prop: 3-NN interp (8192 vs 4096), 42->32 ----
        three_interp<<<gs((long long)B * N, 128), dim3(128), 0, stream>>>(
            pts, l1_pc, l1_new, fea, A, N, NP1, 32, 10, B * N);
        conv_bn_relu(A, prm[FP_W1], prm[FP_G1], prm[FP_B1],
                     (long long)B * N, 42, 32, Bf);

        // ---- bn1 + conv2 head + xyz concat ----
        zero_stats<<<dim3(1), dim3(128), 0, stream>>>(stats, 128);
        bn_stats<<<dim3(256), dim3(256), 0, stream>>>(Bf, stats, stats + 64,
                                                      (long long)B * N, 32);
        final_head<<<gs((long long)B * N, 128), dim3(128), 0, stream>>>(
            Bf, stats, stats + 64, prm[BN1_G], prm[BN1_B],
            prm[CONV2_W], prm[CONV2_B], pts, outp, (long long)B * N, 32);
    };

    float* out = (float*)d_out;
    run_branch(points1, fea1, out);
    run_branch(points2, fea2, out + (size_t)B * N * 15);
}